// ReFINe_plus_28595892257638
// MI455X (gfx1250) — compile-verified
//
#include <hip/hip_runtime.h>

#define EMBED 64

typedef __attribute__((ext_vector_type(2))) float v2f;
typedef __attribute__((ext_vector_type(8))) float v8f;

// ---------------- elementwise / setup kernels ----------------

__global__ void k_fill0(float* __restrict__ p, long n) {
  long i = (long)blockIdx.x * blockDim.x + threadIdx.x;
  if (i < n) p[i] = 0.0f;
}

__global__ void k_deg(const int* __restrict__ col, float* __restrict__ deg, int E) {
  int e = blockIdx.x * blockDim.x + threadIdx.x;
  if (e < E) atomicAdd(&deg[col[e]], 1.0f);
}

__global__ void k_dinv(float* __restrict__ d, int N) {
  int n = blockIdx.x * blockDim.x + threadIdx.x;
  if (n < N) {
    float v = d[n];
    d[n] = (v > 0.0f) ? (1.0f / sqrtf(v)) : 0.0f;
  }
}

__global__ void k_w(const int* __restrict__ row, const int* __restrict__ col,
                    const float* __restrict__ dinv, float* __restrict__ w, int E) {
  int e = blockIdx.x * blockDim.x + threadIdx.x;
  if (e < E) w[e] = dinv[row[e]] * dinv[col[e]];
}

__global__ void k_init_out(const float* __restrict__ emb, const float* __restrict__ alpha,
                           float* __restrict__ outb, long n) {
  long i = (long)blockIdx.x * blockDim.x + threadIdx.x;
  float a0 = alpha[0];            // uniform -> s_load
  if (i < n) outb[i] = emb[i] * a0;
}

__global__ void k_axpy(const float* __restrict__ x, const float* __restrict__ alpha,
                       int ai, float* __restrict__ outb, long n) {
  long i = (long)blockIdx.x * blockDim.x + threadIdx.x;
  float a = alpha[ai];            // uniform -> s_load
  if (i < n) outb[i] += a * x[i];
}

// ---------------- edge scatter: x_new[col] += w * x[row] ----------------
// 16 threads per edge, float4 per thread (64 dims), 4 fp32 atomics per thread.
// All traffic is L2-resident (x/x_new = 20.5 MB each, L2 = 192 MB).

__global__ void k_scatter(const float4* __restrict__ x4,   // [N*16] float4 view of x
                          const int* __restrict__ row, const int* __restrict__ col,
                          const float* __restrict__ w, float* __restrict__ xnew, int E) {
  int t = blockIdx.x * blockDim.x + threadIdx.x;
  int e = t >> 4;
  if (e >= E) return;
  int g = t & 15;
  float4 xv = x4[row[e] * 16 + g];
  float wv = w[e];
  float* dst = xnew + (size_t)col[e] * EMBED + g * 4;
  atomicAdd(dst + 0, wv * xv.x);
  atomicAdd(dst + 1, wv * xv.y);
  atomicAdd(dst + 2, wv * xv.z);
  atomicAdd(dst + 3, wv * xv.w);
}

// ---------------- final stage: 16 dot products per wave via WMMA f32 16x16x4 ----
// C = S(16x64) * D^T(64x16) accumulated as 16 chained V_WMMA_F32_16X16X4_F32;
// result = diag(C). Full fp32, matches reference precision.
//
// A 16x4 layout : lane l holds A[l%16][(l/16)*2 + v], v=0,1   (ISA 7.12.2)
// B 4x16 layout : lane l holds B[(l/16)*2 + v][l%16]
// -> both are float2 loads at out[idx[l%16]]*64 + step*4 + (l/16)*2.
// C 16x16 layout: (M,N) at vgpr M%8, lane (M/8)*16 + N.
// diag(q): lanes 0..7 (vgpr=lane) and lanes 24..31 (vgpr=lane-24, q=lane-16).

__global__ void k_dot_wmma(const float* __restrict__ outb,
                           const int* __restrict__ qsrc, const int* __restrict__ qdst,
                           float* __restrict__ res, int Q) {
  int wave = (blockIdx.x * blockDim.x + threadIdx.x) >> 5;
  int lane = threadIdx.x & 31;
  int qbase = wave * 16;
  if (qbase >= Q) return;              // wave-uniform exit: EXEC stays all-1s for WMMA

  int m = lane & 15;
  int q = qbase + m;
  if (q >= Q) q = Q - 1;               // clamp loads; stores are masked below
  const float* srow = outb + (size_t)qsrc[q] * EMBED;
  const float* drow = outb + (size_t)qdst[q] * EMBED;
  int khalf = (lane >> 4) * 2;         // 0 or 2

  v8f c = {};
#pragma unroll
  for (int s = 0; s < EMBED / 4; ++s) {
    int k = s * 4 + khalf;
    v2f a = *(const v2f*)(srow + k);
    v2f b = *(const v2f*)(drow + k);
    c = __builtin_amdgcn_wmma_f32_16x16x4_f32(
        /*neg_a=*/false, a, /*neg_b=*/false, b,
        /*c_mod=*/(short)0, c, /*reuse_a=*/false, /*reuse_b=*/false);
  }

  // diagonal extraction
  bool writer = false;
  int vsel = 0, qoff = 0;
  if (lane < 8)        { writer = true; vsel = lane;      qoff = lane;      }
  else if (lane >= 24) { writer = true; vsel = lane - 24; qoff = lane - 16; }

  float val = 0.0f;
#pragma unroll
  for (int v = 0; v < 8; ++v) val = (v == vsel) ? c[v] : val;

  if (writer && (qbase + qoff) < Q) res[qbase + qoff] = val;
}

// ---------------- launcher ----------------

extern "C" void kernel_launch(void* const* d_in, const int* in_sizes, int n_in,
                              void* d_out, int out_size, void* d_ws, size_t ws_size,
                              hipStream_t stream) {
  const float* emb   = (const float*)d_in[0];
  const float* alpha = (const float*)d_in[1];
  const int*   eidx  = (const int*)d_in[2];
  const int*   lidx  = (const int*)d_in[3];
  float*       res   = (float*)d_out;

  const int N = in_sizes[0] / EMBED;
  const int L = in_sizes[1] - 1;       // NUM_LAYERS
  const int E = in_sizes[2] / 2;
  const int Q = in_sizes[3] / 2;

  const int* row = eidx;
  const int* col = eidx + E;
  const int* qs  = lidx;
  const int* qd  = lidx + Q;

  const long NV = (long)N * EMBED;

  float* ws   = (float*)d_ws;
  float* dinv = ws;                    // N floats (deg, then dinv in place)
  float* w    = dinv + N;              // E floats
  float* outb = w + E;                 // N*64
  float* x0   = outb + NV;             // N*64
  float* x1   = x0 + NV;               // N*64

  const int B = 256;
  auto nblk = [](long n, int b) { return (int)((n + b - 1) / b); };

  // degrees -> dinv -> edge weights
  k_fill0<<<nblk(N, B), B, 0, stream>>>(dinv, N);
  k_deg  <<<nblk(E, B), B, 0, stream>>>(col, dinv, E);
  k_dinv <<<nblk(N, B), B, 0, stream>>>(dinv, N);
  k_w    <<<nblk(E, B), B, 0, stream>>>(row, col, dinv, w, E);

  // out = alpha[0] * emb
  k_init_out<<<nblk(NV, B), B, 0, stream>>>(emb, alpha, outb, NV);

  // propagation layers (ping-pong x buffers)
  const float* xcur = emb;
  float* bufs[2] = {x0, x1};
  for (int l = 0; l < L; ++l) {
    float* xn = bufs[l & 1];
    k_fill0  <<<nblk(NV, B), B, 0, stream>>>(xn, NV);
    k_scatter<<<nblk((long)E * 16, B), B, 0, stream>>>(
        (const float4*)xcur, row, col, w, xn, E);
    k_axpy   <<<nblk(NV, B), B, 0, stream>>>(xn, alpha, l + 1, outb, NV);
    xcur = xn;
  }

  // final: 16 queries per wave, chained WMMA f32 16x16x4
  long waves = (Q + 15) / 16;
  k_dot_wmma<<<nblk(waves * 32, B), B, 0, stream>>>(outb, qs, qd, res, Q);
}